// BertInsertion_19980187861325
// MI455X (gfx1250) — compile-verified
//
#include <hip/hip_runtime.h>
#include <hip/hip_bf16.h>
#include <math.h>

// ---------------------------------------------------------------------------
// BertInsertion loss on MI455X (gfx1250, wave32).
//   Per batch b: Gram = G * G^T with G = gathered 16x1024 f32 rows, computed
//   with V_WMMA_F32_16X16X4_F32 (256 K-steps). Gram diagonal -> norms,
//   Gram[c][0] -> dot products; then 15-way softmax CE + argmax.
// ---------------------------------------------------------------------------

typedef float v2f __attribute__((ext_vector_type(2)));
typedef float v4f __attribute__((ext_vector_type(4)));
typedef float v8f __attribute__((ext_vector_type(8)));

#define NB    64     // batches
#define NS    512    // sequence length
#define ND    1024   // hidden dim
#define NC    16     // gathered rows
#define EPSV  1e-6f

#define KCHUNK     512              // K staged per LDS pass
#define LDS_STRIDE 516              // 512 + 4 pad; stride % 64 == 4 -> 16 rows hit
                                    // distinct bank groups, conflict-free reads

__global__ __launch_bounds__(32)
void bert_insertion_gram_kernel(const float* __restrict__ seq,
                                const int*   __restrict__ sot,
                                const int*   __restrict__ labels,
                                float*       __restrict__ out,  // [0]=loss, [1..64]=pred, [65..128]=labels
                                float*       __restrict__ ws)   // [0..63]=ce, [64..127]=valid
{
    __shared__ float g[NC * LDS_STRIDE];   // ~33 KB staging tile
    __shared__ int   sidx[NC];
    __shared__ float gram[NC * NC];
    __shared__ float sims[NC - 1];

    const int b    = blockIdx.x;
    const int lane = threadIdx.x;

    // ---- 1) locate the 16 flagged positions (ascending) via wave32 ballot
    int base = 0;
    for (int chunk = 0; chunk < NS / 32; ++chunk) {
        const int s    = chunk * 32 + lane;
        const int flag = sot[b * NS + s];
        const unsigned m = __builtin_amdgcn_ballot_w32(flag != 0);
        if (flag) {
            const int rank = __popc(m & ((1u << lane) - 1u));
            const int slot = base + rank;
            if (slot < NC) sidx[slot] = s;
        }
        base += __popc(m);
    }
    __syncthreads();

    // ---- 2) Gram = G * G^T via f32 WMMA, K staged through LDS in halves
    // A (16x4 f32) lane layout: lane L holds row M=L&15, K-pair 2*(L>>4).
    // B = G^T makes the B-operand fetch identical, so one float2 feeds both.
    v8f acc = {0.f, 0.f, 0.f, 0.f, 0.f, 0.f, 0.f, 0.f};
    const int m_row = lane & 15;
    const int k_sub = (lane >> 4) << 1;   // 0 or 2

    for (int half = 0; half < ND / KCHUNK; ++half) {
        for (int r = 0; r < NC; ++r) {
            const float* src = seq + ((size_t)b * NS + sidx[r]) * ND + half * KCHUNK;
            if (half == 0) __builtin_prefetch(src + KCHUNK, 0, 3); // global_prefetch_b8
            float* dst = &g[r * LDS_STRIDE];
            #pragma unroll
            for (int it = 0; it < KCHUNK / (32 * 4); ++it) {
                const int c = (it * 32 + lane) * 4;
                *(v4f*)(dst + c) = *(const v4f*)(src + c);   // coalesced 512B/iter
            }
        }
        __syncthreads();

        #pragma unroll 8
        for (int k0 = 0; k0 < KCHUNK; k0 += 4) {
            const v2f ab = *(const v2f*)(&g[m_row * LDS_STRIDE + k0 + k_sub]);
            acc = __builtin_amdgcn_wmma_f32_16x16x4_f32(
                      /*neg_a=*/false, ab, /*neg_b=*/false, ab,
                      /*c_mod=*/(short)0, acc,
                      /*reuse_a=*/false, /*reuse_b=*/false);
        }
        __syncthreads();
    }

    // ---- 3) spill Gram (C/D layout: VGPR j -> row j + 8*(lane>=16), col lane&15)
    const int rbase = (lane >> 4) << 3;
    const int col   = lane & 15;
    #pragma unroll
    for (int j = 0; j < 8; ++j)
        gram[(rbase + j) * NC + col] = acc[j];
    __syncthreads();

    // ---- 4) cosine scores for the 15 candidates
    if (lane < NC - 1) {
        const int   c   = lane + 1;
        const float n0  = fmaxf(sqrtf(gram[0]),             EPSV);
        const float nc_ = fmaxf(sqrtf(gram[c * NC + c]),    EPSV);
        sims[lane]      = gram[c * NC + 0] / (n0 * nc_);
    }
    __syncthreads();

    // ---- 5) softmax CE + argmax (15 elements: serial on lane 0 is fine)
    if (lane == 0) {
        float mx = -INFINITY; int arg = 0;
        for (int t = 0; t < NC - 1; ++t) {
            const float v = sims[t];
            if (v > mx) { mx = v; arg = t; }
        }
        float se = 0.f;
        for (int t = 0; t < NC - 1; ++t) se += expf(sims[t] - mx);
        const float lse   = mx + logf(se);
        const int   lab   = labels[b];
        const bool  valid = (lab >= 0);
        const int   slab  = valid ? lab : 0;
        const float ce    = lse - sims[slab];

        ws[b]          = valid ? ce : 0.f;
        ws[NB + b]     = valid ? 1.f : 0.f;
        out[1 + b]     = (float)arg;     // pred_labels
        out[1 + NB + b] = (float)lab;    // labels passthrough
    }
}

__global__ __launch_bounds__(32)
void bert_insertion_reduce_kernel(const float* __restrict__ ws,
                                  float*       __restrict__ out)
{
    const int lane = threadIdx.x;          // 32 lanes cover 64 batches, 2 each
    float ce = ws[lane]          + ws[lane + 32];
    float nv = ws[NB + lane]     + ws[NB + lane + 32];
    #pragma unroll
    for (int off = 16; off > 0; off >>= 1) {
        ce += __shfl_down(ce, off, 32);
        nv += __shfl_down(nv, off, 32);
    }
    if (lane == 0) out[0] = ce / fmaxf(nv, 1.f);
}

extern "C" void kernel_launch(void* const* d_in, const int* in_sizes, int n_in,
                              void* d_out, int out_size, void* d_ws, size_t ws_size,
                              hipStream_t stream)
{
    const float* seq    = (const float*)d_in[0];  // (64, 512, 1024) f32
    const int*   sot    = (const int*)  d_in[1];  // (64, 512) i32
    const int*   labels = (const int*)  d_in[2];  // (64,) i32
    float*       out    = (float*)d_out;          // 129 floats
    float*       ws     = (float*)d_ws;           // >= 128 floats

    bert_insertion_gram_kernel<<<NB, 32, 0, stream>>>(seq, sot, labels, out, ws);
    bert_insertion_reduce_kernel<<<1, 32, 0, stream>>>(ws, out);
}